// SFIS_77077483094379
// MI455X (gfx1250) — compile-verified
//
#include <hip/hip_runtime.h>
#include <math.h>

typedef __attribute__((ext_vector_type(2))) float v2f;
typedef __attribute__((ext_vector_type(8))) float v8f;

#define BATCH 2048
#define NF 16
#define ND 64
#define NH 64
#define P0 120
#define P1 560

// ---------------- Threefry2x32 (exact JAX semantics) ----------------
__host__ __device__ inline unsigned tf_rotl(unsigned v, int r) {
  return (v << r) | (v >> (32 - r));
}
__host__ __device__ inline void threefry2x32(unsigned k0, unsigned k1,
                                             unsigned c0, unsigned c1,
                                             unsigned& o0, unsigned& o1) {
  unsigned ks2 = 0x1BD11BDAu ^ k0 ^ k1;
  unsigned x0 = c0 + k0, x1 = c1 + k1;
#define TFR(r) { x0 += x1; x1 = tf_rotl(x1, r); x1 ^= x0; }
  TFR(13) TFR(15) TFR(26) TFR(6)  x0 += k1;  x1 += ks2 + 1u;
  TFR(17) TFR(29) TFR(16) TFR(24) x0 += ks2; x1 += k0 + 2u;
  TFR(13) TFR(15) TFR(26) TFR(6)  x0 += k0;  x1 += k1 + 3u;
  TFR(17) TFR(29) TFR(16) TFR(24) x0 += k1;  x1 += ks2 + 4u;
  TFR(13) TFR(15) TFR(26) TFR(6)  x0 += ks2; x1 += k0 + 5u;
#undef TFR
  o0 = x0; o1 = x1;
}

__device__ inline float eps_uniform(unsigned k0, unsigned k1, unsigned idx, unsigned ntot) {
  unsigned half = ntot >> 1;            // ntot is even for both levels
  unsigned lo = (idx < half) ? idx : (idx - half);
  unsigned o0, o1;
  threefry2x32(k0, k1, lo, lo + half, o0, o1);
  unsigned bits = (idx < half) ? o0 : o1;
  return __uint_as_float((bits >> 9) | 0x3f800000u) - 1.0f;
}

// ---------------- init ----------------
__global__ void k_init(float* acc2) {
  if (threadIdx.x < 2) acc2[threadIdx.x] = 0.f;
}

// ---------------- gather + M@V / M@Z (WMMA f32 16x16x4) ----------------
__global__ void k_transform(const int* __restrict__ x, const float* __restrict__ M,
                            const float* __restrict__ w0, const float* __restrict__ w_table,
                            const float* __restrict__ V_table, const float* __restrict__ Z_table,
                            float* __restrict__ v0f, float* __restrict__ z0f,
                            float* __restrict__ yacc, float* __restrict__ acc2) {
  __shared__ float sV[NF * ND];
  __shared__ float sZ[NF * ND];
  __shared__ int   sX[NF];
  __shared__ float sW[NF];
  const int b = blockIdx.x;
  const int t = threadIdx.x;              // 64 threads = 2 waves
  if (t < NF) sX[t] = x[b * NF + t];
  __syncthreads();
  for (int f = 0; f < NF; ++f) {
    size_t row = (size_t)sX[f] * ND;
    sV[f * ND + t] = V_table[row + t];
    sZ[f * ND + t] = Z_table[row + t];
  }
  if (t < NF) sW[t] = w_table[sX[t]];
  __syncthreads();
  if (t == 0) {
    float s = 0.f, s2 = 0.f;
    for (int f = 0; f < NF; ++f) { s += sW[f]; s2 += sW[f] * sW[f]; }
    yacc[b] = w0[0] + s;
    atomicAdd(&acc2[0], s2);
  }
  // wave0 -> V0f, wave1 -> Z0f (uniform per wave; EXEC all-1 for WMMA)
  const int lane  = t & 31;
  const float* src = (t < 32) ? sV : sZ;
  float*       dst = (t < 32) ? v0f : z0f;
  const int mrow  = lane & 15;
  const int khalf = (lane >> 4) * 2;
#pragma unroll
  for (int nt = 0; nt < 4; ++nt) {
    v8f c = {};
    const int n = nt * 16 + mrow;
#pragma unroll
    for (int ks = 0; ks < 4; ++ks) {
      const int k = 4 * ks + khalf;
      v2f a;  a.x  = M[mrow * NF + k];      a.y  = M[mrow * NF + k + 1];
      v2f bb; bb.x = src[k * ND + n];       bb.y = src[(k + 1) * ND + n];
      c = __builtin_amdgcn_wmma_f32_16x16x4_f32(false, a, false, bb, (short)0, c, false, false);
    }
#pragma unroll
    for (int v = 0; v < 8; ++v) {
      const int row = v + ((lane >> 4) << 3);
      dst[((size_t)b * NF + row) * ND + n] = c[v];
    }
  }
}

// ---------------- level kernel: products + Z_sel@W1 (WMMA) + relu + @W2 + bernoulli ----------------
template <int LEVEL>
__global__ void k_level(const float* __restrict__ v0f, const float* __restrict__ z0f,
                        const float* __restrict__ W1, const float* __restrict__ b1,
                        const float* __restrict__ W2, const float* __restrict__ b2,
                        float* __restrict__ yacc, float* __restrict__ acc2,
                        unsigned ek0, unsigned ek1) {
  constexpr int P  = (LEVEL == 0) ? P0 : P1;
  constexpr int NT = (P + 15) / 16;      // 8 or 35 tiles
  __shared__ float sW1[ND * NH];
  __shared__ float sB1[NH];
  __shared__ float sW2[NH];
  __shared__ float sB2;
  const int tid = threadIdx.x;           // 256 threads = 8 waves
  for (int idx = tid; idx < ND * NH; idx += 256) sW1[idx] = W1[idx];
  if (tid < NH) { sB1[tid] = b1[tid]; sW2[tid] = W2[tid]; }
  if (tid == 0) sB2 = b2[0];
  __syncthreads();

  const int b     = blockIdx.x;
  const int wave  = tid >> 5;
  const int lane  = tid & 31;
  const int mrow  = lane & 15;
  const int khalf = (lane >> 4) * 2;
  const float* Vb = v0f + (size_t)b * NF * ND;
  const float* Zb = z0f + (size_t)b * NF * ND;

  float yAccum = 0.f, v2Accum = 0.f;

  for (int tI = wave; tI < NT; tI += 8) {
    const int i = tI * 16 + mrow;
    const bool valid = (i < P);
    const int ic = valid ? i : 0;
    int rA, rB, rC = 0; bool kept = false;
    if (LEVEL == 0) {
      int a = 0;
#pragma unroll
      for (int q = 1; q <= 14; ++q) a += (ic >= (q * (q + 1)) / 2);
      const int c = ic - (a * (a + 1)) / 2;
      rA = a + 1; rB = c;
    } else {
      int a = 0;
#pragma unroll
      for (int q = 1; q <= 13; ++q) a += (ic >= (q * (q + 1) * (q + 2)) / 6);
      const int j = ic - (a * (a + 1) * (a + 2)) / 6;   // level-0 kept entry index
      int a0 = 0;
#pragma unroll
      for (int q = 1; q <= 14; ++q) a0 += (j >= (q * (q + 1)) / 2);
      const int c0 = j - (a0 * (a0 + 1)) / 2;
      rA = a + 2; rB = a0 + 1; rC = c0;
      kept = (a <= 12);
    }

    // Build A operand (Z_sel tile) in native WMMA layout; fold Wt & V^2 on the fly.
    v2f az[16];
    float wt = 0.f, v2p = 0.f;
#pragma unroll
    for (int j = 0; j < 16; ++j) {
      const int k = 4 * j + khalf;
      v2f zs = (*(const v2f*)(Zb + rA * ND + k)) * (*(const v2f*)(Zb + rB * ND + k));
      v2f vs = (*(const v2f*)(Vb + rA * ND + k)) * (*(const v2f*)(Vb + rB * ND + k));
      if (LEVEL == 1) {
        zs *= *(const v2f*)(Zb + rC * ND + k);
        vs *= *(const v2f*)(Vb + rC * ND + k);
      }
      if (!valid) { zs.x = 0.f; zs.y = 0.f; vs.x = 0.f; vs.y = 0.f; }
      az[j] = zs;
      wt += vs.x + vs.y;
      if (LEVEL == 1) v2p += kept ? (vs.x * vs.x + vs.y * vs.y) : 0.f;
    }
    wt += __shfl_xor(wt, 16);   // combine k-halves: lanes l and l^16 hold row l&15

    // One threefry per lane per tile: lane (l&15)==m owns eps for row m.
    const float epsv = eps_uniform(ek0, ek1, (unsigned)(b * P + ic), (unsigned)(BATCH * P));

    // GEMM: Z_sel(16x64) @ W1(64x64) via 4 n-tiles x 16 k-steps of wmma f32 16x16x4
    float logitp[8] = {0.f, 0.f, 0.f, 0.f, 0.f, 0.f, 0.f, 0.f};
#pragma unroll
    for (int nt = 0; nt < 4; ++nt) {
      v8f acc = {};
      const int n = nt * 16 + mrow;
#pragma unroll
      for (int j = 0; j < 16; ++j) {
        const int k = 4 * j + khalf;
        v2f bb; bb.x = sW1[k * NH + n]; bb.y = sW1[(k + 1) * NH + n];
        acc = __builtin_amdgcn_wmma_f32_16x16x4_f32(false, az[j], false, bb, (short)0, acc, false, false);
      }
      const float b1n = sB1[n];
      const float w2n = sW2[n];
#pragma unroll
      for (int v = 0; v < 8; ++v) {
        float h = fmaxf(acc[v] + b1n, 0.f);
        logitp[v] = fmaf(h, w2n, logitp[v]);
      }
    }
    // Reduce h@W2 over the 16 lanes of each half (rows v / v+8)
#pragma unroll
    for (int v = 0; v < 8; ++v) {
      logitp[v] += __shfl_xor(logitp[v], 1);
      logitp[v] += __shfl_xor(logitp[v], 2);
      logitp[v] += __shfl_xor(logitp[v], 4);
      logitp[v] += __shfl_xor(logitp[v], 8);
    }
    const float b2v = sB2;
#pragma unroll
    for (int v = 0; v < 8; ++v) {
      const int srcl = (lane < 16) ? v : (v + 24);      // lane holding row v (lo) / v+8 (hi)
      const float wtrow  = __shfl(wt,   srcl);
      const float epsrow = __shfl(epsv, srcl);
      const int row = v + ((lane >> 4) << 3);
      const int ii = tI * 16 + row;
      const float logit = logitp[v] + b2v;
      const float pro = 1.f / (1.f + __expf(-logit));
      const bool fire = ((lane & 15) == 0) && (ii < P) && (pro >= epsrow);
      yAccum += fire ? wtrow : 0.f;
    }
    v2Accum += v2p;
  }

  yAccum += __shfl_xor(yAccum, 16);
  yAccum += __shfl_xor(yAccum, 8);
  yAccum += __shfl_xor(yAccum, 4);
  yAccum += __shfl_xor(yAccum, 2);
  yAccum += __shfl_xor(yAccum, 1);
  if (LEVEL == 1) {
    v2Accum += __shfl_xor(v2Accum, 16);
    v2Accum += __shfl_xor(v2Accum, 8);
    v2Accum += __shfl_xor(v2Accum, 4);
    v2Accum += __shfl_xor(v2Accum, 2);
    v2Accum += __shfl_xor(v2Accum, 1);
  }
  if (lane == 0) {
    atomicAdd(&yacc[b], yAccum);
    if (LEVEL == 1) atomicAdd(&acc2[1], v2Accum);
  }
}

// ---------------- finalize ----------------
__global__ void k_final(const float* __restrict__ yacc, const float* __restrict__ acc2,
                        float* __restrict__ out) {
  const int idx = blockIdx.x * blockDim.x + threadIdx.x;
  if (idx < BATCH) out[idx] = yacc[idx];
  else if (idx == BATCH) out[BATCH] = 1e-4f * (acc2[0] + acc2[1]);
}

extern "C" void kernel_launch(void* const* d_in, const int* in_sizes, int n_in,
                              void* d_out, int out_size, void* d_ws, size_t ws_size,
                              hipStream_t stream) {
  (void)in_sizes; (void)n_in; (void)out_size; (void)ws_size;
  const int*   x       = (const int*)d_in[0];
  const float* M       = (const float*)d_in[1];
  const float* w0      = (const float*)d_in[2];
  const float* w_table = (const float*)d_in[3];
  const float* V_table = (const float*)d_in[4];
  const float* Z_table = (const float*)d_in[5];
  const float* W1      = (const float*)d_in[6];
  const float* b1      = (const float*)d_in[7];
  const float* W2      = (const float*)d_in[8];
  const float* b2      = (const float*)d_in[9];

  float* ws   = (float*)d_ws;
  float* acc2 = ws;                       // [0]=sum w^2, [1]=sum V_keep^2
  float* yacc = ws + 64;                  // BATCH floats
  float* v0f  = ws + 64 + BATCH;          // BATCH*16*64
  float* z0f  = v0f + (size_t)BATCH * NF * ND;

  // eps keys: key(42) = [0,42]; fold_in(key, li) = threefry2x32(key, [0, li])
  unsigned e00, e01, e10, e11;
  threefry2x32(0u, 42u, 0u, 0u, e00, e01);
  threefry2x32(0u, 42u, 0u, 1u, e10, e11);

  k_init<<<1, 32, 0, stream>>>(acc2);
  k_transform<<<BATCH, 64, 0, stream>>>(x, M, w0, w_table, V_table, Z_table,
                                        v0f, z0f, yacc, acc2);
  k_level<0><<<BATCH, 256, 0, stream>>>(v0f, z0f, W1, b1, W2, b2, yacc, acc2, e00, e01);
  k_level<1><<<BATCH, 256, 0, stream>>>(v0f, z0f, W1, b1, W2, b2, yacc, acc2, e10, e11);
  k_final<<<(BATCH + 1 + 255) / 256, 256, 0, stream>>>(yacc, acc2, (float*)d_out);
}